// Cosine_PredictingModule_38122129719601
// MI455X (gfx1250) — compile-verified
//
#include <hip/hip_runtime.h>
#include <math.h>

// ---------------------------------------------------------------------------
// Edge-scoring kernel for MI455X (gfx1250, wave32).
//   out[e] = sigmoid( relu( [h|t|cos] @ W1 + b1 ) @ W2 + b2 )
// GEMM part done with V_WMMA_F32_16X16X4_F32 (exact fp32, matches reference).
// One wave handles a 16-edge tile: M=16 edges, N=64 (4 tiles), K=128 in 32
// WMMA steps; cosine feature added as rank-1 epilogue.
// ---------------------------------------------------------------------------

typedef float v2f __attribute__((ext_vector_type(2)));
typedef float v8f __attribute__((ext_vector_type(8)));

#define EMBED            64
#define WAVES_PER_BLOCK  8
#define BLOCK            (WAVES_PER_BLOCK * 32)
#define TILE_EDGES       16
#define EDGES_PER_BLOCK  (WAVES_PER_BLOCK * TILE_EDGES)
// W1 staged transposed in LDS: row n (output channel), 128 K entries, padded
// to stride 132 floats -> the wave's ds_load_b64 pattern (16 low lanes at
// kb=4s, 16 high lanes at kb=4s+2) covers all 64 LDS banks exactly once.
#define W1T_STRIDE       132

__global__ __launch_bounds__(BLOCK)
void edge_mlp_wmma_kernel(const float* __restrict__ hc,   // [N_CUST, 64]
                          const float* __restrict__ hp,   // [N_PROD, 64]
                          const int*   __restrict__ src,  // [E]
                          const int*   __restrict__ dst,  // [E]
                          const float* __restrict__ W1,   // [129, 64]
                          const float* __restrict__ b1,   // [64]
                          const float* __restrict__ W2,   // [64]
                          const float* __restrict__ b2,   // [1]
                          float*       __restrict__ out,  // [E]
                          int E)
{
    __shared__ float sW1t[64 * W1T_STRIDE]; // W1[0:128,:] transposed
    __shared__ float sW1cos[64];            // W1[128,:]  (cosine feature row)
    __shared__ float sB1[64];
    __shared__ float sW2[64];

    const int tid = threadIdx.x;

    // Cooperative stage of weights (coalesced global reads).
    for (int i = tid; i < 129 * 64; i += BLOCK) {
        const int k = i >> 6;
        const int n = i & 63;
        const float v = W1[i];
        if (k < 128) sW1t[n * W1T_STRIDE + k] = v;
        else         sW1cos[n] = v;
    }
    for (int i = tid; i < 64; i += BLOCK) { sB1[i] = b1[i]; sW2[i] = W2[i]; }
    __syncthreads();

    const int lane = tid & 31;
    const int wave = tid >> 5;
    const int nl   = lane & 15;   // column-in-tile / row-in-tile selector
    const int hi   = lane >> 4;   // wave half (K interleave per ISA layout)
    const int edgeBase = (blockIdx.x * WAVES_PER_BLOCK + wave) * TILE_EDGES;
    if (edgeBase >= E) return;    // wave-uniform: EXEC stays all-1s for WMMA

    // Each lane pair (L, L+16) serves edge M = L&15. Clamp keeps lanes active
    // on a partial tail tile (stores are guarded below).
    int e = edgeBase + nl;
    if (e >= E) e = E - 1;
    const float* __restrict__ hrow = hc + (size_t)src[e] * EMBED;
    const float* __restrict__ trow = hp + (size_t)dst[e] * EMBED;

    v8f acc[4] = {};      // D tiles: columns n0 = 0,16,32,48
    v2f hbuf[16];         // head fragments, reused for the fused dot product
    float hh = 0.f, tt = 0.f, dp = 0.f;

    // ---- Phase 1: K = 0..63 (head features), buffer A fragments ----------
#pragma unroll
    for (int s = 0; s < 16; ++s) {
        const int kb = 4 * s + 2 * hi;          // ISA 16x4 f32 A layout
        v2f a;
        a.x = hrow[kb];
        a.y = hrow[kb + 1];
        hbuf[s] = a;
        hh += a.x * a.x + a.y * a.y;
#pragma unroll
        for (int t = 0; t < 4; ++t) {
            const float* wp = &sW1t[(t * 16 + nl) * W1T_STRIDE + kb];
            v2f b; b.x = wp[0]; b.y = wp[1];
            acc[t] = __builtin_amdgcn_wmma_f32_16x16x4_f32(
                false, a, false, b, (short)0, acc[t], false, false);
        }
    }

    // ---- Phase 2: K = 64..127 (tail features), fuse cosine partials ------
#pragma unroll
    for (int s = 0; s < 16; ++s) {
        const int kb = 4 * s + 2 * hi;
        v2f a;
        a.x = trow[kb];
        a.y = trow[kb + 1];
        tt += a.x * a.x + a.y * a.y;
        dp += hbuf[s].x * a.x + hbuf[s].y * a.y;
        const int kw = 64 + kb;
#pragma unroll
        for (int t = 0; t < 4; ++t) {
            const float* wp = &sW1t[(t * 16 + nl) * W1T_STRIDE + kw];
            v2f b; b.x = wp[0]; b.y = wp[1];
            acc[t] = __builtin_amdgcn_wmma_f32_16x16x4_f32(
                false, a, false, b, (short)0, acc[t], false, false);
        }
    }

    // ---- Cosine (reference-faithful eps handling) ------------------------
    hh += __shfl_xor(hh, 16, 32);
    tt += __shfl_xor(tt, 16, 32);
    dp += __shfl_xor(dp, 16, 32);
    const float nh  = sqrtf(hh);
    const float nt  = sqrtf(tt);
    const float inh = 1.f / fmaxf(nh, 1e-12f);
    const float itn = 1.f / fmaxf(nt, 1e-12f);
    const float den = fmaxf((nh * inh) * (nt * itn), 1e-8f);
    const float cosv = (dp * inh * itn) / den;   // lane L holds cos of edge L&15

    // ---- Epilogue: +b1, +cos*W1[128], ReLU, layer 2, sigmoid -------------
    float w1c[4], b1v[4], w2v[4];
#pragma unroll
    for (int t = 0; t < 4; ++t) {
        const int n = t * 16 + nl;
        w1c[t] = sW1cos[n];
        b1v[t] = sB1[n];
        w2v[t] = sW2[n];
    }
    const float b2s = b2[0];

    float r8[8];
#pragma unroll
    for (int r = 0; r < 8; ++r) {
        const int m = r + 8 * hi;                 // D layout: VGPR r -> row M
        const float cm = __shfl(cosv, m, 32);     // cos of edge m lives in lane m
        float p = 0.f;
#pragma unroll
        for (int t = 0; t < 4; ++t) {
            float x = acc[t][r] + b1v[t] + cm * w1c[t];
            x = fmaxf(x, 0.f);                    // ReLU
            p += x * w2v[t];                      // x @ W2 partial
        }
        p += __shfl_xor(p, 1, 32);                // reduce over 16 N-lanes
        p += __shfl_xor(p, 2, 32);
        p += __shfl_xor(p, 4, 32);
        p += __shfl_xor(p, 8, 32);
        r8[r] = p;
    }

    if (nl == 0) {                                // lanes 0 and 16 write 8 each
#pragma unroll
        for (int r = 0; r < 8; ++r) {
            const int eo = edgeBase + r + 8 * hi;
            if (eo < E) {
                const float z = r8[r] + b2s;
                out[eo] = 1.f / (1.f + expf(-z));
            }
        }
    }
}

extern "C" void kernel_launch(void* const* d_in, const int* in_sizes, int n_in,
                              void* d_out, int out_size, void* d_ws, size_t ws_size,
                              hipStream_t stream)
{
    const float* hc  = (const float*)d_in[0];
    const float* hp  = (const float*)d_in[1];
    const int*   src = (const int*)  d_in[2];
    const int*   dst = (const int*)  d_in[3];
    const float* W1  = (const float*)d_in[4];
    const float* b1  = (const float*)d_in[5];
    const float* W2  = (const float*)d_in[6];
    const float* b2  = (const float*)d_in[7];
    float* out = (float*)d_out;

    const int E = in_sizes[2];
    const int grid = (E + EDGES_PER_BLOCK - 1) / EDGES_PER_BLOCK;

    hipLaunchKernelGGL(edge_mlp_wmma_kernel, dim3(grid), dim3(BLOCK), 0, stream,
                       hc, hp, src, dst, W1, b1, W2, b2, out, E);
}